// LengthRegulator_7344394076708
// MI455X (gfx1250) — compile-verified
//
#include <hip/hip_runtime.h>
#include <hip/hip_bf16.h>

// LengthRegulator for MI455X (gfx1250).
// Memory-bound gather: out[b,m,:] = x[b, idx(b,m), :]; ~125MB streaming stores
// is the roofline (~5-6us @ 23.3 TB/s); x (32MB) stays L2-resident (192MB L2).
// No matmul -> no WMMA. CDNA5 path used: ASYNCcnt global<->LDS b128 copies
// (data never touches VGPRs), batched 4 frames per block for MLP.

#define LR_B 32
#define LR_T 512
#define LR_D 512
#define LR_FPB 4   // frames per block

// ---------------------------------------------------------------------------
// Kernel 1: per-row inclusive cumsum of durations (with clamp-to->=0).
// One block per batch row, T threads, Hillis-Steele scan in LDS.
// 32 blocks total: perf noise, keep simple/correct.
// ---------------------------------------------------------------------------
__global__ __launch_bounds__(LR_T) void lr_cumsum_kernel(
    const int* __restrict__ dur, int* __restrict__ tot) {
  __shared__ int s[LR_T];
  const int b = blockIdx.x;
  const int t = threadIdx.x;

  int v = dur[b * LR_T + t];
  v = v < 0 ? 0 : v;                 // duration[duration < 0] = 0
  s[t] = v;
  __syncthreads();

#pragma unroll
  for (int off = 1; off < LR_T; off <<= 1) {
    int u = (t >= off) ? s[t - off] : 0;
    __syncthreads();
    s[t] += u;                       // only thread t writes s[t]
    __syncthreads();
  }
  tot[b * LR_T + t] = s[t];
}

// ---------------------------------------------------------------------------
// Kernel 2: one block per 4 consecutive output frames (b, m0..m0+3).
//   1) lane-parallel predicated binary search: lane f (mod 4) finds the first
//      token j with cumsum[j] > m0+f; fixed 10 iterations, no divergence,
//      clamped mid so no OOB. Results broadcast via v_readlane.
//   2) 4 async global->LDS b128 loads in flight, one s_wait_asynccnt 0, then
//      4 async LDS->global b128 stores. Each thread moves 16B per frame.
// Block = 128 threads (4 wave32s); LDS staging = 4 * 2KB = 8KB.
// Each wave touches only its own LDS slices: no barrier needed.
// ---------------------------------------------------------------------------
__global__ __launch_bounds__(128) void lr_expand_kernel(
    const float* __restrict__ x, const int* __restrict__ tot,
    float* __restrict__ out, int M) {
  const int m0  = blockIdx.x * LR_FPB;   // first output frame of this block
  const int b   = blockIdx.y;            // batch row
  const int tid = threadIdx.x;

  const int* row = tot + b * LR_T;

  // --- lane-parallel binary search (searchsorted side='right') -------------
  // Lane l searches for frame m0 + (l & 3). Fully predicated, fixed trip
  // count (512 -> ... -> 1 needs 10 halvings), mid clamped so the load is
  // always in-bounds even after lo==hi.
  const int mq = m0 + (tid & 3);
  int lo = 0, hi = LR_T;
#pragma unroll
  for (int it = 0; it < 10; ++it) {
    int mid  = (lo + hi) >> 1;
    int safe = mid < LR_T ? mid : LR_T - 1;
    int v    = row[safe];
    bool go  = lo < hi;
    int nlo  = (v <= mq) ? mid + 1 : lo;
    int nhi  = (v <= mq) ? hi      : mid;
    lo = go ? nlo : lo;
    hi = go ? nhi : hi;
  }

  // --- async copy: 4 loads in flight, one wait, 4 stores -------------------
  __shared__ __attribute__((aligned(16))) unsigned char lds_buf[LR_FPB * 2048];
  // Low 32 bits of the generic address of a __shared__ object == LDS byte
  // offset (hardware adds LDS_BASE; aperture bits live in the high half).
  const unsigned lds_base =
      (unsigned)(unsigned long long)(void*)&lds_buf[0];

#pragma unroll
  for (int f = 0; f < LR_FPB; ++f) {
    int idxf = __builtin_amdgcn_readlane(lo, f);     // frame m0+f's token
    idxf = (idxf < LR_T - 1) ? idxf : (LR_T - 1);    // trailing-fill clamp
    if (m0 + f < M) {                                 // uniform tail guard
      const unsigned long long gsrc =
          (unsigned long long)(x + ((size_t)b * LR_T + idxf) * LR_D) +
          (unsigned long long)tid * 16ull;
      const unsigned lds_addr = lds_base + (unsigned)f * 2048u +
                                (unsigned)tid * 16u;
      // copies 16 bytes from global address gsrc to LDS address lds_addr (tracked by ASYNCcnt)
      asm volatile("global_load_async_to_lds_b128 %0, %1, off"
                   :
                   : "v"(lds_addr), "v"(gsrc)
                   : "memory");
    }
  }

  // Wait for this wave's async loads to land in LDS.
  asm volatile("s_wait_asynccnt 0x0" ::: "memory");

#pragma unroll
  for (int f = 0; f < LR_FPB; ++f) {
    if (m0 + f < M) {                                 // uniform tail guard
      const unsigned long long gdst =
          (unsigned long long)(out + ((size_t)b * M + (m0 + f)) * LR_D) +
          (unsigned long long)tid * 16ull;
      const unsigned lds_addr = lds_base + (unsigned)f * 2048u +
                                (unsigned)tid * 16u;
      // copies 16 bytes from LDS address lds_addr to global address gdst (tracked by ASYNCcnt)
      // s_endpgm performs an implicit wait-idle: no trailing wait needed.
      asm volatile("global_store_async_from_lds_b128 %0, %1, off"
                   :
                   : "v"(gdst), "v"(lds_addr)
                   : "memory");
    }
  }
}

// ---------------------------------------------------------------------------
// Host-side launcher. Inputs per reference setup_inputs():
//   d_in[0] = x   : float32 [B, T, D]
//   d_in[1] = dur : integer [B, T]   (harness passes integers as int32)
// d_out: float32 [B, M, D] with M = out_size / (B*D).
// d_ws : tot cumsum scratch, B*T ints = 64KB.
// ---------------------------------------------------------------------------
extern "C" void kernel_launch(void* const* d_in, const int* in_sizes, int n_in,
                              void* d_out, int out_size, void* d_ws, size_t ws_size,
                              hipStream_t stream) {
  (void)in_sizes; (void)n_in; (void)ws_size;

  const float* x   = (const float*)d_in[0];
  const int*   dur = (const int*)d_in[1];
  float*       out = (float*)d_out;
  int*         tot = (int*)d_ws;

  const int M = out_size / (LR_B * LR_D);

  lr_cumsum_kernel<<<dim3(LR_B), dim3(LR_T), 0, stream>>>(dur, tot);
  if (M > 0) {
    const int nblk = (M + LR_FPB - 1) / LR_FPB;
    lr_expand_kernel<<<dim3(nblk, LR_B), dim3(128), 0, stream>>>(x, tot, out, M);
  }
}